// PointSample_22943715295830
// MI455X (gfx1250) — compile-verified
//
#include <hip/hip_runtime.h>
#include <cstdint>
#include <cstddef>

// Bilinear point sample (grid_sample, align_corners=False, zero padding).
// features: (B=8, H=128, W=128, C=256) f32   grid: (B, 16384, 2) f32
// out: (B, 16384, 256) f32
//
// Memory-bound gather -> CDNA5 async global->LDS copies (ASYNCcnt) with
// per-wave double buffering. One wave32 per point per iteration:
//   - 8x global_load_async_to_lds_b128 (SADDR form, 32-bit row offsets)
//     stage the 4 corner rows (4KB) into LDS
//   - s_wait_asynccnt pipelines point i+1's loads behind point i's blend
//   - each lane blends 8 channels (2x float4) and stores 32B coalesced.

#define PS_H 128
#define PS_W 128
#define PS_C 256
#define WAVES_PER_BLOCK 8
#define PTS_PER_WAVE 4

struct Wts { float w[4]; };

// One corner row (1KB): two async b128 transfers. The instruction's imm
// offset is added to BOTH the LDS and global addresses, so one (vdst, voff)
// pair covers both 512B halves.
__device__ __forceinline__ void ps_async_row(uint32_t ldsDst, uint32_t goff,
                                             const float* __restrict__ feat) {
    asm volatile("global_load_async_to_lds_b128 %0, %1, %2"
                 :: "v"(ldsDst), "v"(goff), "s"(feat) : "memory");
    asm volatile("global_load_async_to_lds_b128 %0, %1, %2 offset:512"
                 :: "v"(ldsDst), "v"(goff), "s"(feat) : "memory");
}

// Compute corner offsets + weights for point p, issue 8 async loads into the
// LDS buffer at byte offset ldsBuf. All offsets are 32-bit (tensor = 128MiB).
__device__ __forceinline__ Wts ps_issue_point(const float* __restrict__ features,
                                              const float* __restrict__ grid,
                                              int p, uint32_t laneB, uint32_t ldsBuf) {
    const uint32_t batchOff = (uint32_t)(p >> 14) << 24;   // b * H*W*C*4
    const float* g = grid + (size_t)p * 2;
    float x = g[0] * (float)PS_W - 0.5f;                   // grid[...,0] -> W
    float y = g[1] * (float)PS_H - 0.5f;                   // grid[...,1] -> H
    float fx0 = floorf(x), fy0 = floorf(y);
    int ix0 = (int)fx0, iy0 = (int)fy0;
    float fx = x - fx0, fy = y - fy0;

    Wts wt;
    wt.w[0] = (1.0f - fy) * (1.0f - fx);                   // (iy0,   ix0)
    wt.w[1] = fy * (1.0f - fx);                            // (iy0+1, ix0)
    wt.w[2] = (1.0f - fy) * fx;                            // (iy0,   ix0+1)
    wt.w[3] = fy * fx;                                     // (iy0+1, ix0+1)

    const int iy[4] = { iy0, iy0 + 1, iy0,     iy0 + 1 };
    const int ix[4] = { ix0, ix0,     ix0 + 1, ix0 + 1 };

#pragma unroll
    for (int k = 0; k < 4; ++k) {
        // zero-pad semantics: out-of-range corner contributes 0 -> weight 0,
        // clamp the address in-range (clamped data is finite, x0 = 0).
        bool ok = ((unsigned)iy[k] < (unsigned)PS_H) & ((unsigned)ix[k] < (unsigned)PS_W);
        wt.w[k] = ok ? wt.w[k] : 0.0f;
        int yc = min(max(iy[k], 0), PS_H - 1);
        int xc = min(max(ix[k], 0), PS_W - 1);
        uint32_t rowOff = batchOff + ((uint32_t)yc << 17) + ((uint32_t)xc << 10);
        ps_async_row(ldsBuf + (uint32_t)(k * 1024) + laneB, rowOff + laneB, features);
    }
    return wt;
}

__device__ __forceinline__ void ps_blend_store(const float* __restrict__ lbuf,
                                               Wts wt, float* __restrict__ outp, int lane) {
    const float w0 = wt.w[0], w1 = wt.w[1], w2 = wt.w[2], w3 = wt.w[3];
    const int co = lane * 8;                               // this lane's 8 channels
    const float4 r0a = *(const float4*)(lbuf + 0 * PS_C + co);
    const float4 r0b = *(const float4*)(lbuf + 0 * PS_C + co + 4);
    const float4 r1a = *(const float4*)(lbuf + 1 * PS_C + co);
    const float4 r1b = *(const float4*)(lbuf + 1 * PS_C + co + 4);
    const float4 r2a = *(const float4*)(lbuf + 2 * PS_C + co);
    const float4 r2b = *(const float4*)(lbuf + 2 * PS_C + co + 4);
    const float4 r3a = *(const float4*)(lbuf + 3 * PS_C + co);
    const float4 r3b = *(const float4*)(lbuf + 3 * PS_C + co + 4);

    float4 oa, ob;
    oa.x = fmaf(w0, r0a.x, fmaf(w1, r1a.x, fmaf(w2, r2a.x, w3 * r3a.x)));
    oa.y = fmaf(w0, r0a.y, fmaf(w1, r1a.y, fmaf(w2, r2a.y, w3 * r3a.y)));
    oa.z = fmaf(w0, r0a.z, fmaf(w1, r1a.z, fmaf(w2, r2a.z, w3 * r3a.z)));
    oa.w = fmaf(w0, r0a.w, fmaf(w1, r1a.w, fmaf(w2, r2a.w, w3 * r3a.w)));
    ob.x = fmaf(w0, r0b.x, fmaf(w1, r1b.x, fmaf(w2, r2b.x, w3 * r3b.x)));
    ob.y = fmaf(w0, r0b.y, fmaf(w1, r1b.y, fmaf(w2, r2b.y, w3 * r3b.y)));
    ob.z = fmaf(w0, r0b.z, fmaf(w1, r1b.z, fmaf(w2, r2b.z, w3 * r3b.z)));
    ob.w = fmaf(w0, r0b.w, fmaf(w1, r1b.w, fmaf(w2, r2b.w, w3 * r3b.w)));

    float4* o = (float4*)(outp + co);
    o[0] = oa;
    o[1] = ob;
}

__global__ __launch_bounds__(256) void PointSample_22943715295830_kernel(
    const float* __restrict__ features, const float* __restrict__ grid,
    float* __restrict__ out, int npoints) {
    // Per-wave double buffer: 2 x (4 rows x 256 f32) = 8KB; 8 waves -> 64KB.
    __shared__ __align__(16) float smem[WAVES_PER_BLOCK][2][4 * PS_C];

    const int lane = threadIdx.x & 31;
    const uint32_t laneB = (uint32_t)lane * 16u;           // 16B chunk per lane
    const int wv = threadIdx.x >> 5;
    // Low 32 bits of the generic LDS address are the LDS byte offset.
    const uint32_t ldsBase = (uint32_t)(uintptr_t)(void*)&smem[wv][0][0];

    const int wave = blockIdx.x * WAVES_PER_BLOCK + wv;
    const int nwaves = gridDim.x * WAVES_PER_BLOCK;

    int p = wave;
    int cur = 0;
    Wts wcur = {};
    if (p < npoints) wcur = ps_issue_point(features, grid, p, laneB, ldsBase);

    while (p < npoints) {
        const int pn = p + nwaves;
        const bool hasNext = pn < npoints;
        Wts wnext = {};
        if (hasNext) {
            // Drain in-flight LDS reads before the async engine rewrites the
            // other buffer (LDS WAR).
            asm volatile("s_wait_dscnt 0x0" ::: "memory");
            wnext = ps_issue_point(features, grid, pn, laneB,
                                   ldsBase + (uint32_t)((cur ^ 1) * 4096));
        }
        // Async loads complete in order: <=8 outstanding means the current
        // point's 8 transfers have landed in LDS.
        if (hasNext) asm volatile("s_wait_asynccnt 0x8" ::: "memory");
        else         asm volatile("s_wait_asynccnt 0x0" ::: "memory");

        ps_blend_store(&smem[wv][cur][0], wcur, out + (size_t)p * PS_C, lane);

        wcur = wnext;
        p = pn;
        cur ^= 1;
    }
}

extern "C" void kernel_launch(void* const* d_in, const int* in_sizes, int n_in,
                              void* d_out, int out_size, void* d_ws, size_t ws_size,
                              hipStream_t stream) {
    const float* features = (const float*)d_in[0];
    const float* grid = (const float*)d_in[1];
    float* out = (float*)d_out;

    const int npoints = in_sizes[1] / 2;   // (B, 16384, 2) -> B*16384 points
    const int ptsPerBlock = WAVES_PER_BLOCK * PTS_PER_WAVE;
    int blocks = (npoints + ptsPerBlock - 1) / ptsPerBlock;
    if (blocks < 1) blocks = 1;

    PointSample_22943715295830_kernel<<<blocks, 256, 0, stream>>>(
        features, grid, out, npoints);
}